// GraphAttentionLayer_89335319757572
// MI455X (gfx1250) — compile-verified
//
#include <hip/hip_runtime.h>
#include <math.h>

typedef float v2f __attribute__((ext_vector_type(2)));
typedef float v8f __attribute__((ext_vector_type(8)));

#define B_     32
#define N_     1024
#define F_     256
#define M_     (B_*N_)     /* 32768 rows (b,i) or (b,n) flattened */
#define EPS_   1e-5f
#define ALPHA_ 0.2f

// ---------------------------------------------------------------------------
// GEMM1: Wh[m,o] = sum_f h[m,f] * W[o,f]   (M=32768, N=256, K=256)
// block = 512 thr (16 waves). Tile: 16 rows x 256 cols. A staged in LDS.
// ---------------------------------------------------------------------------
__global__ __launch_bounds__(512)
void gat_gemm1_wh(const float* __restrict__ h, const float* __restrict__ W,
                  float* __restrict__ Wh) {
  __shared__ float At[16 * 257];              // pitch 257 -> conflict-free col reads
  const int tid  = threadIdx.x;
  const int row0 = blockIdx.x * 16;
  for (int i = tid; i < 16 * 256; i += 512) {
    int r = i >> 8, c = i & 255;
    At[r * 257 + c] = h[(size_t)(row0 + r) * F_ + c];
  }
  __syncthreads();
  const int wave = tid >> 5;
  const int lid  = tid & 31;
  const int half = lid >> 4;   // which half-wave (K split)
  const int ln   = lid & 15;   // m for A, n for B/D
  const int o0   = wave * 16;
  v8f acc = {};
  for (int kb = 0; kb < 256; kb += 4) {
    const int k = kb + 2 * half;
    v2f a, b;
    a.x = At[ln * 257 + k];
    a.y = At[ln * 257 + k + 1];
    // B[k][n] = W[o0+n][k]  (W is row-major [F,F])
    b.x = W[(size_t)(o0 + ln) * F_ + k];
    b.y = W[(size_t)(o0 + ln) * F_ + k + 1];
    acc = __builtin_amdgcn_wmma_f32_16x16x4_f32(false, a, false, b,
                                                (short)0, acc, false, false);
  }
  for (int r = 0; r < 8; ++r) {
    const int m = row0 + r + 8 * half;
    Wh[(size_t)m * F_ + o0 + ln] = acc[r];
  }
}

// ---------------------------------------------------------------------------
// ei[m] = Wh[m,:].a1 ; ej[m] = Wh[m,:].a2   (wave-per-row)
// ---------------------------------------------------------------------------
__global__ __launch_bounds__(256)
void gat_ei_ej(const float* __restrict__ Wh, const float* __restrict__ a1,
               const float* __restrict__ a2, float* __restrict__ ei,
               float* __restrict__ ej) {
  const int wave = threadIdx.x >> 5;
  const int lane = threadIdx.x & 31;
  const int row  = blockIdx.x * 8 + wave;
  const float* p = Wh + (size_t)row * F_;
  float s1 = 0.f, s2 = 0.f;
  for (int f = lane; f < F_; f += 32) {
    const float v = p[f];
    s1 += v * a1[f];
    s2 += v * a2[f];
  }
  for (int off = 16; off; off >>= 1) {
    s1 += __shfl_down(s1, off, 32);
    s2 += __shfl_down(s2, off, 32);
  }
  if (lane == 0) { ei[row] = s1; ej[row] = s2; }
}

// ---------------------------------------------------------------------------
// bn1 stats per j over all (b,i): att = ei[b,i] + ej[b,j]
// ---------------------------------------------------------------------------
__global__ __launch_bounds__(256)
void gat_bn1_stats(const float* __restrict__ ei, const float* __restrict__ ej,
                   float* __restrict__ mu1, float* __restrict__ inv1) {
  __shared__ float rs[256], rq[256];
  const int j = blockIdx.x;
  float s = 0.f, q = 0.f;
  for (int smp = threadIdx.x; smp < M_; smp += 256) {
    const float v = ei[smp] + ej[(smp & ~(N_ - 1)) | j];
    s += v; q += v * v;
  }
  rs[threadIdx.x] = s; rq[threadIdx.x] = q;
  __syncthreads();
  for (int off = 128; off; off >>= 1) {
    if (threadIdx.x < off) {
      rs[threadIdx.x] += rs[threadIdx.x + off];
      rq[threadIdx.x] += rq[threadIdx.x + off];
    }
    __syncthreads();
  }
  if (threadIdx.x == 0) {
    const float mean = rs[0] * (1.f / (float)M_);
    const float var  = rq[0] * (1.f / (float)M_) - mean * mean;
    mu1[j]  = mean;
    inv1[j] = rsqrtf(var + EPS_);
  }
}

// ---------------------------------------------------------------------------
// bn2 stats per j over (b,i) of leaky(bn1(att))
// ---------------------------------------------------------------------------
__global__ __launch_bounds__(256)
void gat_bn2_stats(const float* __restrict__ ei, const float* __restrict__ ej,
                   const float* __restrict__ mu1, const float* __restrict__ inv1,
                   const float* __restrict__ g1, const float* __restrict__ b1,
                   float* __restrict__ mu2, float* __restrict__ inv2) {
  __shared__ float rs[256], rq[256];
  const int j = blockIdx.x;
  const float c1 = inv1[j] * g1[j];
  const float m1 = mu1[j];
  const float bb = b1[j];
  float s = 0.f, q = 0.f;
  for (int smp = threadIdx.x; smp < M_; smp += 256) {
    const float z = (ei[smp] + ej[(smp & ~(N_ - 1)) | j] - m1) * c1 + bb;
    const float y = z >= 0.f ? z : ALPHA_ * z;
    s += y; q += y * y;
  }
  rs[threadIdx.x] = s; rq[threadIdx.x] = q;
  __syncthreads();
  for (int off = 128; off; off >>= 1) {
    if (threadIdx.x < off) {
      rs[threadIdx.x] += rs[threadIdx.x + off];
      rq[threadIdx.x] += rq[threadIdx.x + off];
    }
    __syncthreads();
  }
  if (threadIdx.x == 0) {
    const float mean = rs[0] * (1.f / (float)M_);
    const float var  = rq[0] * (1.f / (float)M_) - mean * mean;
    mu2[j]  = mean;
    inv2[j] = rsqrtf(var + EPS_);
  }
}

// ---------------------------------------------------------------------------
// Softmax stats per row (b,i): rowmax over j, rowinv = 1/sum(exp(w-max))
// ---------------------------------------------------------------------------
__global__ __launch_bounds__(256)
void gat_softmax_stats(const float* __restrict__ ei, const float* __restrict__ ej,
                       const float* __restrict__ mu1, const float* __restrict__ inv1,
                       const float* __restrict__ g1, const float* __restrict__ b1,
                       const float* __restrict__ mu2, const float* __restrict__ inv2,
                       const float* __restrict__ g2, const float* __restrict__ b2,
                       float* __restrict__ rowmax, float* __restrict__ rowinv) {
  __shared__ float wbuf[N_];
  __shared__ float red[256];
  const int rowi = blockIdx.x;           // b*N + i
  const int base = rowi & ~(N_ - 1);     // b*N
  const float e_i = ei[rowi];
  float mx = -3.4e38f;
  for (int j = threadIdx.x; j < N_; j += 256) {
    const float z = (e_i + ej[base + j] - mu1[j]) * (inv1[j] * g1[j]) + b1[j];
    const float y = z >= 0.f ? z : ALPHA_ * z;
    const float w = (y - mu2[j]) * (inv2[j] * g2[j]) + b2[j];
    wbuf[j] = w;
    mx = fmaxf(mx, w);
  }
  red[threadIdx.x] = mx;
  __syncthreads();
  for (int off = 128; off; off >>= 1) {
    if (threadIdx.x < off)
      red[threadIdx.x] = fmaxf(red[threadIdx.x], red[threadIdx.x + off]);
    __syncthreads();
  }
  mx = red[0];
  __syncthreads();
  float s = 0.f;
  for (int j = threadIdx.x; j < N_; j += 256) s += __expf(wbuf[j] - mx);
  red[threadIdx.x] = s;
  __syncthreads();
  for (int off = 128; off; off >>= 1) {
    if (threadIdx.x < off) red[threadIdx.x] += red[threadIdx.x + off];
    __syncthreads();
  }
  if (threadIdx.x == 0) { rowmax[rowi] = mx; rowinv[rowi] = 1.f / red[0]; }
}

// ---------------------------------------------------------------------------
// GEMM2: hp[b,i,o] = sum_j att[b,i,j] * Wh[b,j,o], att generated on the fly.
// grid = (N/16, B), block = 512 (16 waves -> 16 rows x 256 cols output tile).
// ---------------------------------------------------------------------------
__global__ __launch_bounds__(512)
void gat_gemm2_hp(const float* __restrict__ Wh, const float* __restrict__ adj,
                  const float* __restrict__ ei, const float* __restrict__ ej,
                  const float* __restrict__ mu1, const float* __restrict__ inv1,
                  const float* __restrict__ g1, const float* __restrict__ b1,
                  const float* __restrict__ mu2, const float* __restrict__ inv2,
                  const float* __restrict__ g2, const float* __restrict__ b2,
                  const float* __restrict__ rowmax, const float* __restrict__ rowinv,
                  float* __restrict__ hp) {
  __shared__ float At[16 * 65];            // 16 x 64 att chunk, pitch 65
  const int tid  = threadIdx.x;
  const int b    = blockIdx.y;
  const int i0   = blockIdx.x * 16;
  const int wave = tid >> 5;
  const int lid  = tid & 31;
  const int half = lid >> 4;
  const int ln   = lid & 15;
  const int o0   = wave * 16;
  const size_t rowBase = (size_t)b * N_;
  v8f acc = {};
  for (int jb = 0; jb < N_; jb += 64) {
    // cooperative on-the-fly A-tile: softmax-normalized masked attention
    for (int idx = tid; idx < 16 * 64; idx += 512) {
      const int r  = idx >> 6, c = idx & 63;
      const int j  = jb + c;
      const int gi = (int)rowBase + i0 + r;     // b*N + i
      const float z = (ei[gi] + ej[rowBase + j] - mu1[j]) * (inv1[j] * g1[j]) + b1[j];
      const float y = z >= 0.f ? z : ALPHA_ * z;
      const float w = (y - mu2[j]) * (inv2[j] * g2[j]) + b2[j];
      const float p = __expf(w - rowmax[gi]) * rowinv[gi];
      At[r * 65 + c] = p * adj[(size_t)gi * N_ + j];
    }
    __syncthreads();
    for (int kk = 0; kk < 64; kk += 4) {
      const int k = kk + 2 * half;
      v2f a, bb;
      a.x = At[ln * 65 + k];
      a.y = At[ln * 65 + k + 1];
      const size_t jrow = rowBase + jb + k;
      bb.x = Wh[jrow * F_ + o0 + ln];
      bb.y = Wh[(jrow + 1) * F_ + o0 + ln];
      acc = __builtin_amdgcn_wmma_f32_16x16x4_f32(false, a, false, bb,
                                                  (short)0, acc, false, false);
    }
    __syncthreads();
  }
  for (int r = 0; r < 8; ++r) {
    const size_t m = rowBase + i0 + r + 8 * half;
    hp[m * F_ + o0 + ln] = acc[r];
  }
}

// ---------------------------------------------------------------------------
// bn3 stats per feature o over all (b,i)
// ---------------------------------------------------------------------------
__global__ __launch_bounds__(256)
void gat_bn3_stats(const float* __restrict__ hp, float* __restrict__ mu3,
                   float* __restrict__ inv3) {
  __shared__ float rs[256], rq[256];
  const int o = blockIdx.x;
  float s = 0.f, q = 0.f;
  for (int row = threadIdx.x; row < M_; row += 256) {
    const float v = hp[(size_t)row * F_ + o];
    s += v; q += v * v;
  }
  rs[threadIdx.x] = s; rq[threadIdx.x] = q;
  __syncthreads();
  for (int off = 128; off; off >>= 1) {
    if (threadIdx.x < off) {
      rs[threadIdx.x] += rs[threadIdx.x + off];
      rq[threadIdx.x] += rq[threadIdx.x + off];
    }
    __syncthreads();
  }
  if (threadIdx.x == 0) {
    const float mean = rs[0] * (1.f / (float)M_);
    const float var  = rq[0] * (1.f / (float)M_) - mean * mean;
    mu3[o]  = mean;
    inv3[o] = rsqrtf(var + EPS_);
  }
}

// ---------------------------------------------------------------------------
// out = elu(bn3(hp))
// ---------------------------------------------------------------------------
__global__ __launch_bounds__(256)
void gat_bn3_elu(const float* __restrict__ hp, const float* __restrict__ mu3,
                 const float* __restrict__ inv3, const float* __restrict__ g3,
                 const float* __restrict__ b3, float* __restrict__ out) {
  const size_t g = (size_t)blockIdx.x * 256 + threadIdx.x;
  const int o = (int)(g & (F_ - 1));
  const float x = (hp[g] - mu3[o]) * (inv3[o] * g3[o]) + b3[o];
  out[g] = x >= 0.f ? x : expm1f(x);
}

// ---------------------------------------------------------------------------
extern "C" void kernel_launch(void* const* d_in, const int* in_sizes, int n_in,
                              void* d_out, int out_size, void* d_ws, size_t ws_size,
                              hipStream_t stream) {
  (void)in_sizes; (void)n_in; (void)out_size; (void)ws_size;
  const float* h   = (const float*)d_in[0];
  const float* adj = (const float*)d_in[1];
  const float* W   = (const float*)d_in[2];
  const float* a1  = (const float*)d_in[3];
  const float* a2  = (const float*)d_in[4];
  const float* g1  = (const float*)d_in[5];
  const float* b1  = (const float*)d_in[6];
  const float* g2  = (const float*)d_in[7];
  const float* b2  = (const float*)d_in[8];
  const float* g3  = (const float*)d_in[9];
  const float* b3  = (const float*)d_in[10];
  float* out = (float*)d_out;

  float* ws   = (float*)d_ws;
  float* Wh   = ws;                   // 8388608
  float* hp   = Wh + (size_t)M_ * F_; // 8388608
  float* ei   = hp + (size_t)M_ * F_; // 32768
  float* ej   = ei + M_;              // 32768
  float* rmax = ej + M_;              // 32768
  float* rinv = rmax + M_;            // 32768
  float* mu1  = rinv + M_;            // 1024
  float* inv1 = mu1 + N_;
  float* mu2  = inv1 + N_;
  float* inv2 = mu2 + N_;
  float* mu3  = inv2 + N_;            // 256
  float* inv3 = mu3 + F_;

  gat_gemm1_wh<<<M_ / 16, 512, 0, stream>>>(h, W, Wh);
  gat_ei_ej<<<M_ / 8, 256, 0, stream>>>(Wh, a1, a2, ei, ej);
  gat_bn1_stats<<<N_, 256, 0, stream>>>(ei, ej, mu1, inv1);
  gat_bn2_stats<<<N_, 256, 0, stream>>>(ei, ej, mu1, inv1, g1, b1, mu2, inv2);
  gat_softmax_stats<<<M_, 256, 0, stream>>>(ei, ej, mu1, inv1, g1, b1,
                                            mu2, inv2, g2, b2, rmax, rinv);
  dim3 g2grid(N_ / 16, B_);
  gat_gemm2_hp<<<g2grid, 512, 0, stream>>>(Wh, adj, ei, ej, mu1, inv1, g1, b1,
                                           mu2, inv2, g2, b2, rmax, rinv, hp);
  gat_bn3_stats<<<F_, 256, 0, stream>>>(hp, mu3, inv3);
  gat_bn3_elu<<<(M_ * F_) / 256, 256, 0, stream>>>(hp, mu3, inv3, g3, b3, out);
}